// MultiresolutionHashEncoding_80625126081121
// MI455X (gfx1250) — compile-verified
//
#include <hip/hip_runtime.h>
#include <cstdint>
#include <cmath>

#define N_LEVELS      16
#define LOG2_HASH     19
#define HASHMAP_SIZE  (1u << LOG2_HASH)
#define HASH_MASK     (HASHMAP_SIZE - 1u)
#define PRIME1        2654435761u
#define PRIME2        805459861u
#define BLOCK         256

typedef __attribute__((ext_vector_type(4))) float v4f;

struct LevelParams {
    int   res[N_LEVELS];
    float resf[N_LEVELS];
    int   dense[N_LEVELS];
};

#if defined(__has_builtin)
#if __has_builtin(__builtin_amdgcn_global_load_async_to_lds_b32) && \
    __has_builtin(__builtin_amdgcn_s_wait_asynccnt)
#define USE_ASYNC_LDS 1
#endif
#endif

__global__ __launch_bounds__(BLOCK) void
hashgrid_encode_kernel(const float* __restrict__ positions,
                       const float* __restrict__ embeddings,
                       float* __restrict__ out,
                       LevelParams lp, int n_points)
{
    __shared__ float s_pos[3 * BLOCK];

    const int tid        = threadIdx.x;
    const int block_base = blockIdx.x * BLOCK;
    const int point      = block_base + tid;

    // ---- Stage this block's positions through LDS (coalesced b32 loads) ----
    const float* gsrc = positions + (size_t)block_base * 3;
    const int    limit = n_points * 3 - block_base * 3;

#ifdef USE_ASYNC_LDS
    #pragma unroll
    for (int k = 0; k < 3; ++k) {
        int i = tid + k * BLOCK;
        if (i < limit) {
            // CDNA5 async DMA: global -> LDS without touching VGPRs (ASYNCcnt)
            __builtin_amdgcn_global_load_async_to_lds_b32(
                (__attribute__((address_space(1))) int*)(uintptr_t)(gsrc + i),
                (__attribute__((address_space(3))) int*)(&s_pos[i]),
                0, 0);
        }
    }
    __builtin_amdgcn_s_wait_asynccnt(0);
    __syncthreads();
#else
    #pragma unroll
    for (int k = 0; k < 3; ++k) {
        int i = tid + k * BLOCK;
        if (i < limit) s_pos[i] = gsrc[i];
    }
    __syncthreads();
#endif

    if (point >= n_points) return;

    const float px = s_pos[3 * tid + 0];
    const float py = s_pos[3 * tid + 1];
    const float pz = s_pos[3 * tid + 2];

    float out32[2 * N_LEVELS];

    #pragma unroll
    for (int l = 0; l < N_LEVELS; ++l) {
        const float rf = lp.resf[l];

        const float sx = px * rf, sy = py * rf, sz = pz * rf;
        const float fx = floorf(sx), fy = floorf(sy), fz = floorf(sz);
        const int   vx = (int)fx,   vy = (int)fy,   vz = (int)fz;
        const float wx = sx - fx,   wy = sy - fy,   wz = sz - fz;

        uint32_t i0, i1, i2, i3, i4, i5, i6, i7;
        if (lp.dense[l]) {                       // scalar (SGPR) branch per level
            const uint32_t r1   = (uint32_t)lp.res[l] + 1u;
            const uint32_t r1sq = r1 * r1;
            const uint32_t base = (uint32_t)vx * r1sq + (uint32_t)vy * r1 + (uint32_t)vz;
            i0 = base;              i1 = base + 1u;
            i2 = base + r1;         i3 = base + r1 + 1u;
            i4 = base + r1sq;       i5 = base + r1sq + 1u;
            i6 = base + r1sq + r1;  i7 = base + r1sq + r1 + 1u;
        } else {
            const uint32_t hx0 = (uint32_t)vx;
            const uint32_t hx1 = (uint32_t)(vx + 1);
            const uint32_t hy0 = (uint32_t)vy       * PRIME1;
            const uint32_t hy1 = (uint32_t)(vy + 1) * PRIME1;
            const uint32_t hz0 = (uint32_t)vz       * PRIME2;
            const uint32_t hz1 = (uint32_t)(vz + 1) * PRIME2;
            i0 = (hx0 ^ hy0 ^ hz0) & HASH_MASK;  // (0,0,0)
            i1 = (hx0 ^ hy0 ^ hz1) & HASH_MASK;  // (0,0,1)
            i2 = (hx0 ^ hy1 ^ hz0) & HASH_MASK;  // (0,1,0)
            i3 = (hx0 ^ hy1 ^ hz1) & HASH_MASK;  // (0,1,1)
            i4 = (hx1 ^ hy0 ^ hz0) & HASH_MASK;  // (1,0,0)
            i5 = (hx1 ^ hy0 ^ hz1) & HASH_MASK;  // (1,0,1)
            i6 = (hx1 ^ hy1 ^ hz0) & HASH_MASK;  // (1,1,0)
            i7 = (hx1 ^ hy1 ^ hz1) & HASH_MASK;  // (1,1,1)
        }

        // 8 x global_load_b64 gathers; tables are L2-resident (64MB < 192MB L2)
        const float2* __restrict__ tab =
            (const float2*)(embeddings + (size_t)l * (HASHMAP_SIZE * 2));
        const float2 f0 = tab[i0], f1 = tab[i1], f2 = tab[i2], f3 = tab[i3];
        const float2 f4 = tab[i4], f5 = tab[i5], f6 = tab[i6], f7 = tab[i7];

        // Trilinear interpolation (matches reference corner ordering c = i*4+j*2+k)
        const float c00x = f0.x + wx * (f4.x - f0.x), c00y = f0.y + wx * (f4.y - f0.y);
        const float c01x = f1.x + wx * (f5.x - f1.x), c01y = f1.y + wx * (f5.y - f1.y);
        const float c10x = f2.x + wx * (f6.x - f2.x), c10y = f2.y + wx * (f6.y - f2.y);
        const float c11x = f3.x + wx * (f7.x - f3.x), c11y = f3.y + wx * (f7.y - f3.y);

        const float c0x = c00x + wy * (c10x - c00x), c0y = c00y + wy * (c10y - c00y);
        const float c1x = c01x + wy * (c11x - c01x), c1y = c01y + wy * (c11y - c01y);

        out32[2 * l + 0] = c0x + wz * (c1x - c0x);
        out32[2 * l + 1] = c0y + wz * (c1y - c0y);
    }

    // ---- Streaming output: 8 x 128-bit NON-TEMPORAL stores (don't evict the
    //      L2-resident hash tables with 256MB of write-once output) ----
    v4f* __restrict__ o4 = (v4f*)(out + (size_t)point * (2 * N_LEVELS));
    #pragma unroll
    for (int i = 0; i < (2 * N_LEVELS) / 4; ++i) {
        v4f v = { out32[4 * i + 0], out32[4 * i + 1],
                  out32[4 * i + 2], out32[4 * i + 3] };
        __builtin_nontemporal_store(v, o4 + i);
    }
}

extern "C" void kernel_launch(void* const* d_in, const int* in_sizes, int n_in,
                              void* d_out, int out_size, void* d_ws, size_t ws_size,
                              hipStream_t stream) {
    const float* positions  = (const float*)d_in[0];
    const float* embeddings = (const float*)d_in[1];
    float*       out        = (float*)d_out;

    const int n_points = in_sizes[0] / 3;

    // Per-level resolutions computed with the reference's exact double-precision
    // formula (floor boundaries like level 15 -> 2048 are numerically delicate).
    LevelParams lp;
    const double scale = exp((log(2048.0) - log(16.0)) / 15.0);
    for (int l = 0; l < N_LEVELS; ++l) {
        const int r = (int)floor(16.0 * pow(scale, (double)l));
        lp.res[l]  = r;
        lp.resf[l] = (float)r;
        const long long r1 = (long long)r + 1;
        lp.dense[l] = (r1 * r1 * r1 <= (long long)HASHMAP_SIZE) ? 1 : 0;
    }

    const int grid = (n_points + BLOCK - 1) / BLOCK;
    hashgrid_encode_kernel<<<grid, BLOCK, 0, stream>>>(positions, embeddings, out,
                                                       lp, n_points);
}